// MLPPredictor_34720515621375
// MI455X (gfx1250) — compile-verified
//
#include <hip/hip_runtime.h>
#include <stdint.h>

typedef __attribute__((ext_vector_type(2))) float v2f;
typedef __attribute__((ext_vector_type(8))) float v8f;
typedef int gv4i __attribute__((vector_size(16)));   // matches builtin's pointee type

#define D_FEAT      128
#define TWO_D       256
#define OUTC        2
#define TILE_M      16
#define WAVES_PB    2
#define THREADS     (WAVES_PB * 32)
#define LDS_STRIDE  260   // 256 + 4 pad: 260 % 64 == 4 -> conflict-free 16-row column reads

#if __has_builtin(__builtin_amdgcn_global_load_async_to_lds_b128)
#define HAVE_ASYNC_LDS 1
#else
#define HAVE_ASYNC_LDS 0
#endif

__global__ __launch_bounds__(THREADS)
void edge_mlp_wmma_f32(const float* __restrict__ h,      // [N_NODES][128]
                       const int*   __restrict__ src,    // [E]
                       const int*   __restrict__ dst,    // [E]
                       const float* __restrict__ W,      // [2][256]
                       const float* __restrict__ bias,   // [2]
                       float*       __restrict__ out,    // [E][2]
                       int nEdges, int nTiles)
{
    __shared__ __align__(16) float Wlds[3][LDS_STRIDE];                 // row 2 = zeros (pad lanes)
    __shared__ __align__(16) float tileA[WAVES_PB][TILE_M][LDS_STRIDE]; // combined [16][256] per wave

    const int tid  = threadIdx.x;
    const int wave = tid >> 5;
    const int lane = tid & 31;
    const int nloc = lane & 15;   // N (for B/D) and M (for A) within half-wave
    const int hi   = lane >> 4;   // 0: lanes 0-15 (K=0,1), 1: lanes 16-31 (K=2,3)

    // ---- stage W into LDS rows 0..1, zero row 2 (pad lanes read zeros -> branchless B)
    #pragma unroll
    for (int i = tid; i < 3 * LDS_STRIDE; i += THREADS) {
        int r = i / LDS_STRIDE, c = i - r * LDS_STRIDE;
        Wlds[r][c] = (r < OUTC && c < TWO_D) ? W[r * TWO_D + c] : 0.0f;
    }

    const int  tile  = blockIdx.x * WAVES_PB + wave;
    const bool valid = (tile < nTiles);
    const int  t0    = (valid ? tile : 0) * TILE_M;

    // ---- per-lane node index: lane r<16 holds src row r, lane r>=16 holds dst row r-16
    int eIdx = t0 + nloc;
    if (eIdx >= nEdges) eIdx = nEdges - 1;
    int myRow = (lane < 16) ? src[eIdx] : dst[eIdx];

    // ---- stage 32 rows (16 edges x {src,dst}) into LDS; one 512B coalesced burst per row
    #pragma unroll 4
    for (int r = 0; r < 32; ++r) {
        int rowNode = __shfl(myRow, r);                         // wave32 broadcast
        const float* gp = h + (long long)rowNode * D_FEAT + lane * 4;
        float*       lp = &tileA[wave][r & 15][((r >> 4) ? D_FEAT : 0) + lane * 4];
#if HAVE_ASYNC_LDS
        // CDNA5 direct-to-LDS async copy (ASYNCcnt tracked, no VGPR round-trip)
        __builtin_amdgcn_global_load_async_to_lds_b128(
            (__attribute__((address_space(1))) gv4i*)gp,
            (__attribute__((address_space(3))) gv4i*)lp, 0, 0);
#else
        *(float4*)lp = *(const float4*)gp;
#endif
    }
#if HAVE_ASYNC_LDS
#if __has_builtin(__builtin_amdgcn_s_wait_asynccnt)
    __builtin_amdgcn_s_wait_asynccnt(0);
#else
    asm volatile("s_wait_asynccnt 0x0" ::: "memory");
#endif
#endif
    __syncthreads();

    // ---- D(16x16) = A(16x256) x B(256x16, cols>=2 zero), K in chunks of 4 via WMMA f32
    v8f acc = {};
    const float* arow = &tileA[wave][nloc][hi * 2];
    const float* brow = &Wlds[(nloc < OUTC) ? nloc : 2][hi * 2];
    #pragma unroll 8
    for (int kc = 0; kc < TWO_D / 4; ++kc) {
        v2f a = *(const v2f*)(arow + kc * 4);
        v2f b = *(const v2f*)(brow + kc * 4);
        // (neg_a, A, neg_b, B, c_mod, C, reuse_a, reuse_b)
        acc = __builtin_amdgcn_wmma_f32_16x16x4_f32(false, a, false, b,
                                                    (short)0, acc, false, false);
    }

    // ---- store: lanes with N<2 hold 8 row results each (VGPR r -> M = hi*8 + r)
    if (nloc < OUTC) {
        const float bn = bias[nloc];
        const int mbase = hi * 8;
        if (valid && (t0 + TILE_M <= nEdges)) {
            // fast path: full tile, unconditional stores
            const long long base = (long long)(t0 + mbase) * OUTC + nloc;
            #pragma unroll
            for (int r = 0; r < 8; ++r)
                out[base + (long long)r * OUTC] = acc[r] + bn;
        } else if (valid) {
            #pragma unroll
            for (int r = 0; r < 8; ++r) {
                int e = t0 + mbase + r;
                if (e < nEdges) out[(long long)e * OUTC + nloc] = acc[r] + bn;
            }
        }
    }
}

extern "C" void kernel_launch(void* const* d_in, const int* in_sizes, int n_in,
                              void* d_out, int out_size, void* d_ws, size_t ws_size,
                              hipStream_t stream) {
    const float* h    = (const float*)d_in[0];
    const int*   src  = (const int*)  d_in[1];
    const int*   dst  = (const int*)  d_in[2];
    const float* W    = (const float*)d_in[3];
    const float* bias = (const float*)d_in[4];
    float*       out  = (float*)d_out;

    const int nEdges = in_sizes[1];
    const int nTiles = (nEdges + TILE_M - 1) / TILE_M;
    const int blocks = (nTiles + WAVES_PB - 1) / WAVES_PB;

    edge_mlp_wmma_f32<<<blocks, THREADS, 0, stream>>>(h, src, dst, W, bias, out,
                                                      nEdges, nTiles);
}